// Angles2BasisDihedral_46858093199454
// MI455X (gfx1250) — compile-verified
//
#include <hip/hip_runtime.h>

// ---------------------------------------------------------------------------
// Angles2BasisDihedral for MI455X (gfx1250, wave32).
// One wave per batch row:
//   Phase 0: async global->LDS staging (global_load_async_to_lds_b128).
//   Phase 1: per-lane serial 16-residue segment product; sincos cached in regs.
//   Phase 2: wave-wide Hillis-Steele scan of 3x3 matrix products (EXEC-guarded
//            combine: measured better codegen than per-element selects).
//   Phase 3: re-expand segment from cached sincos, masked b96 stores.
// Store-bandwidth bound (~302 MB out -> ~13 us floor @ 23.3 TB/s).
// WMMA deliberately unused: the only f32 WMMA (16x16x4) cannot pack
// independent 3x3 chain steps; marshalling into striped layouts costs more
// than the 27-FMA in-lane product it would replace.
// ---------------------------------------------------------------------------

#define WAVE 32

struct M3 { float m[9]; };
struct F3 { float x, y, z; };

__device__ __forceinline__ M3 m3_identity() {
  M3 r;
  r.m[0] = 1.f; r.m[1] = 0.f; r.m[2] = 0.f;
  r.m[3] = 0.f; r.m[4] = 1.f; r.m[5] = 0.f;
  r.m[6] = 0.f; r.m[7] = 0.f; r.m[8] = 1.f;
  return r;
}

// c = a @ b (general)
__device__ __forceinline__ M3 mmul(const M3& a, const M3& b) {
  M3 c;
#pragma unroll
  for (int i = 0; i < 3; ++i) {
#pragma unroll
    for (int j = 0; j < 3; ++j) {
      c.m[i * 3 + j] = fmaf(a.m[i * 3 + 0], b.m[0 + j],
                       fmaf(a.m[i * 3 + 1], b.m[3 + j],
                            a.m[i * 3 + 2] * b.m[6 + j]));
    }
  }
  return c;
}

// c = g @ R(alpha,beta) where R = [[ca, -sa*cb,  sa*sb],
//                                  [sa,  ca*cb, -ca*sb],
//                                  [ 0,     sb,     cb]]
// Exploits R[2][0]==0 and shares the four pairwise products across rows.
__device__ __forceinline__ M3 mmul_rot(const M3& g, float sa, float ca,
                                       float sb, float cb) {
  const float sacb = sa * cb, cacb = ca * cb;
  const float sasb = sa * sb, casb = ca * sb;
  M3 c;
#pragma unroll
  for (int i = 0; i < 3; ++i) {
    const float g0 = g.m[i * 3 + 0], g1 = g.m[i * 3 + 1], g2 = g.m[i * 3 + 2];
    c.m[i * 3 + 0] = fmaf(g0, ca, g1 * sa);
    c.m[i * 3 + 1] = fmaf(g0, -sacb, fmaf(g1, cacb, g2 * sb));
    c.m[i * 3 + 2] = fmaf(g0, sasb, fmaf(g1, -casb, g2 * cb));
  }
  return c;
}

__device__ __forceinline__ void store_row(float* p, float a, float b, float c) {
  F3 v; v.x = a; v.y = b; v.z = c;
  *reinterpret_cast<F3*>(p) = v;   // global_store_b96
}

// ---------------------------------------------------------------------------
// Specialized kernel: one wave32 workgroup per batch row; L must be a
// compile-time multiple of 64.
// ---------------------------------------------------------------------------
template <int L>
__global__ __launch_bounds__(WAVE) void angles2basis_wave(
    const float* __restrict__ inp,   // (B, 2, L)
    const int* __restrict__ alen,    // (B,)
    float* __restrict__ out) {       // (B, 3, 3*(L+1))
  constexpr int S  = L / WAVE;       // residues per lane
  constexpr int RS = 3 * (L + 1);    // output row stride (floats)

  __shared__ float s_ab[2 * L];      // [0,L): alpha, [L,2L): beta

  const int b    = blockIdx.x;
  const int lane = threadIdx.x;

  // ---- Phase 0: async-stage this batch's 2*L floats into LDS ------------
  {
    const float* src = inp + (size_t)b * (2 * L);
    unsigned long long sbase = (unsigned long long)(uintptr_t)src;
    unsigned vbyte = (unsigned)lane * 16u;                       // per-lane offset
    unsigned lbyte = (unsigned)(uintptr_t)(&s_ab[0]) + vbyte;    // LDS byte addr
#pragma unroll
    for (int it = 0; it < (2 * L * 4) / (WAVE * 16); ++it) {
      asm volatile("global_load_async_to_lds_b128 %0, %1, %2"
                   :
                   : "v"(lbyte), "v"(vbyte), "s"(sbase)
                   : "memory");
      vbyte += WAVE * 16u;
      lbyte += WAVE * 16u;
    }
    asm volatile("s_wait_asynccnt 0" ::: "memory");
  }

  const int base = lane * S;

  // ---- Phase 1: per-lane segment product, caching sincos in registers ----
  float sa[S], ca[S], sb[S], cb[S];
  M3 T = m3_identity();
#pragma unroll
  for (int r = 0; r < S; ++r) {
    __sincosf(s_ab[base + r],     &sa[r], &ca[r]);
    __sincosf(s_ab[L + base + r], &sb[r], &cb[r]);
    T = mmul_rot(T, sa[r], ca[r], sb[r], cb[r]);  // r==0 folds (T==I)
  }

  // ---- Phase 2: wave-wide inclusive scan (non-commutative matmul) --------
  // T_l <- T_{l-d} @ T_l keeps left-to-right chain order. EXEC-guarded
  // combine: 2 SALU ops beat 9 per-element selects per step on gfx1250.
#pragma unroll
  for (int d = 1; d < WAVE; d <<= 1) {
    M3 U;
#pragma unroll
    for (int k = 0; k < 9; ++k) U.m[k] = __shfl_up(T.m[k], (unsigned)d, WAVE);
    if (lane >= d) T = mmul(U, T);
  }
  // exclusive prefix P_l = T_{l-1}, P_0 = I
  M3 P;
#pragma unroll
  for (int k = 0; k < 9; ++k) P.m[k] = __shfl_up(T.m[k], 1u, WAVE);
  if (lane == 0) P = m3_identity();

  // ---- Phase 3: re-expand segment from cached sincos, emit prefixes ------
  const int n = alen[b];
  float* ob = out + (size_t)b * (3 * (size_t)RS);

  if (lane == 0) {  // basis[0] = I (index 0 is never masked: n >= 0)
    store_row(ob + 0 * RS, 1.f, 0.f, 0.f);
    store_row(ob + 1 * RS, 0.f, 1.f, 0.f);
    store_row(ob + 2 * RS, 0.f, 0.f, 1.f);
  }

  M3 G = P;
#pragma unroll
  for (int r = 0; r < S; ++r) {
    G = mmul_rot(G, sa[r], ca[r], sb[r], cb[r]);
    const int lo = base + r + 1;            // basis index in [1, L]
    const bool keep = (lo <= n);
#pragma unroll
    for (int i = 0; i < 3; ++i) {
      store_row(ob + (size_t)i * RS + lo * 3,
                keep ? G.m[i * 3 + 0] : 0.f,
                keep ? G.m[i * 3 + 1] : 0.f,
                keep ? G.m[i * 3 + 2] : 0.f);
    }
  }
}

// ---------------------------------------------------------------------------
// Generic fallback: one thread per batch row, serial chain.
// ---------------------------------------------------------------------------
__global__ void angles2basis_generic(const float* __restrict__ inp,
                                     const int* __restrict__ alen,
                                     float* __restrict__ out,
                                     int B, int L) {
  int b = blockIdx.x * blockDim.x + threadIdx.x;
  if (b >= B) return;
  const float* al = inp + (size_t)b * 2 * L;
  const float* be = al + L;
  const int RS = 3 * (L + 1);
  float* ob = out + (size_t)b * 3 * (size_t)RS;
  const int n = alen[b];

  store_row(ob + 0 * RS, 1.f, 0.f, 0.f);
  store_row(ob + 1 * RS, 0.f, 1.f, 0.f);
  store_row(ob + 2 * RS, 0.f, 0.f, 1.f);

  M3 G = m3_identity();
  for (int l = 0; l < L; ++l) {
    float sa, ca, sb, cb;
    __sincosf(al[l], &sa, &ca);
    __sincosf(be[l], &sb, &cb);
    G = mmul_rot(G, sa, ca, sb, cb);
    const int lo = l + 1;
    const bool keep = (lo <= n);
    for (int i = 0; i < 3; ++i) {
      store_row(ob + (size_t)i * RS + lo * 3,
                keep ? G.m[i * 3 + 0] : 0.f,
                keep ? G.m[i * 3 + 1] : 0.f,
                keep ? G.m[i * 3 + 2] : 0.f);
    }
  }
}

// ---------------------------------------------------------------------------
extern "C" void kernel_launch(void* const* d_in, const int* in_sizes, int n_in,
                              void* d_out, int out_size, void* d_ws, size_t ws_size,
                              hipStream_t stream) {
  const float* inp  = (const float*)d_in[0];
  const int*   alen = (const int*)d_in[1];
  float*       out  = (float*)d_out;

  const int B = in_sizes[1];
  const int L = (int)((long long)in_sizes[0] / (2LL * B));

  if (L == 512) {
    angles2basis_wave<512><<<dim3(B), dim3(WAVE), 0, stream>>>(inp, alen, out);
  } else {
    const int thr = 128;
    angles2basis_generic<<<dim3((B + thr - 1) / thr), dim3(thr), 0, stream>>>(
        inp, alen, out, B, L);
  }
}